// Attention_52836687675794
// MI455X (gfx1250) — compile-verified
//
#include <hip/hip_runtime.h>
#include <hip/hip_bf16.h>
#include <math.h>

typedef __attribute__((ext_vector_type(2))) float v2f;
typedef __attribute__((ext_vector_type(8))) float v8f;

#define TX   30
#define DIN  128
#define DH   10
#define NS   64
#define DA   64                 // 2*N_A
#define ROW_STRIDE (TX * DA)    // 1920 floats per batch row of `a`
#define ROWS_PER_WAVE  16
#define WAVES_PER_BLOCK 8
#define ROWS_PER_BLOCK 128

// Dynamic-LDS layout (float offsets).
// W1 stored transposed+padded: W1s[t][m][k] = W1[t][k][m] for m<10, else 0.
//   m-stride = 132 (= 4 mod 64 banks -> conflict-free b64 lane pattern)
#define M_STRIDE   132
#define T_STRIDE   (16 * M_STRIDE)            // 2112 floats per timestep
#define OFF_W1 0
#define W1_ELEMS (TX * T_STRIDE)              // 63360
#define OFF_B1 (OFF_W1 + W1_ELEMS)            // 63360 : b1 padded to 16 cols
#define OFF_W2 (OFF_B1 + TX * 16)             // 63840 : W2 padded to 16 cols
#define OFF_B2 (OFF_W2 + TX * 16)             // 64320
#define OFF_E  (OFF_B2 + 32)                  // 64352 : e[wave][row(16)][t(30)]
#define LDS_FLOATS (OFF_E + WAVES_PER_BLOCK * ROWS_PER_WAVE * TX)  // 68192 (~273 KB)

__device__ __forceinline__ float xor16_sum(float x) {
    // SWAPX16: group-of-32 ds_swizzle, xor_mask=0x10, and_mask=0x1f
    int other = __builtin_amdgcn_ds_swizzle(__float_as_int(x), 0x401F);
    return x + __int_as_float(other);
}

__global__ void __launch_bounds__(256)
attn_wmma_kernel(const float* __restrict__ s, const float* __restrict__ a,
                 const float* __restrict__ W1, const float* __restrict__ b1,
                 const float* __restrict__ W2, const float* __restrict__ b2,
                 float* __restrict__ out)
{
    extern __shared__ float lds[];
    const int tid = threadIdx.x;

    // ---- Stage weights into LDS: transpose + zero-pad so the hot loop has
    //      NO conditionals (all predication cost paid once here) ----
    for (int i = tid; i < TX * 16 * DIN; i += 256) {
        int t  = i >> 11;          // / 2048
        int rm = i & 2047;
        int m  = rm >> 7;          // / 128
        int k  = rm & 127;
        float v = (m < DH) ? W1[(t * DIN + k) * DH + m] : 0.0f;
        lds[OFF_W1 + t * T_STRIDE + m * M_STRIDE + k] = v;
    }
    for (int i = tid; i < TX * 16; i += 256) {
        int t = i >> 4, h = i & 15;
        float bb = (h < DH) ? b1[t * DH + h] : 0.0f;
        float ww = (h < DH) ? W2[t * DH + h] : 0.0f;
        lds[OFF_B1 + i] = bb;
        lds[OFF_W2 + i] = ww;
    }
    if (tid < TX) lds[OFF_B2 + tid] = b2[tid];
    __syncthreads();

    const int wave = tid >> 5;
    const int lane = tid & 31;
    const int rowL = lane & 15;        // N index (batch row) / M index (=h) for A
    const int hi   = lane >> 4;        // which K-pair within a 16x16x4 fragment
    const int r    = blockIdx.x * ROWS_PER_BLOCK + wave * ROWS_PER_WAVE + rowL;

    const float* __restrict__ arow = a + (size_t)r * ROW_STRIDE;
    float* e_rows = lds + OFF_E + wave * (ROWS_PER_WAVE * TX);

    // ---- Preload t-invariant s-half of the B fragments (K = 0..63) ----
    // B(4x16) fragment: v0 = B[kb+2*hi][n=rowL], v1 = B[kb+2*hi+1][n]
    v2f bs[16];
    {
        const float* srow = s + (size_t)r * NS;
        #pragma unroll
        for (int kk = 0; kk < 16; ++kk) {
            int k = kk * 4 + 2 * hi;
            bs[kk] = *(const v2f*)(srow + k);
        }
    }

    // ---- Per-timestep: z = tanh(concat @ W1[t] + b1[t]); e = relu(z.W2[t] + b2[t]) ----
    for (int t = 0; t < TX; ++t) {
        // a-half of B fragments (K = 64..127)
        v2f ba[16];
        const float* at = arow + t * DA;
        #pragma unroll
        for (int kk = 0; kk < 16; ++kk) {
            int k = kk * 4 + 2 * hi;
            ba[kk] = *(const v2f*)(at + k);
        }

        // A fragments: W1s[t][m=rowL][k], two consecutive k -> one ds_load_b64
        const float* w1row = lds + OFF_W1 + t * T_STRIDE + rowL * M_STRIDE;
        v2f av[32];
        #pragma unroll
        for (int kk = 0; kk < 32; ++kk) {
            int k = kk * 4 + 2 * hi;
            av[kk] = *(const v2f*)(w1row + k);
        }

        v8f acc = {0.f, 0.f, 0.f, 0.f, 0.f, 0.f, 0.f, 0.f};
        #pragma unroll
        for (int kk = 0; kk < 32; ++kk) {
            v2f Bv = (kk < 16) ? bs[kk] : ba[kk - 16];
            acc = __builtin_amdgcn_wmma_f32_16x16x4_f32(
                false, av[kk], false, Bv, (short)0, acc, false, false);
        }

        // D[m=h][n=row]: lane holds h = v + 8*hi for its row (n = rowL).
        // Padded h>=10: acc=0, b1=0, W2=0  ->  tanh(0)*0 = 0, no conditionals.
        float ep = 0.0f;
        #pragma unroll
        for (int v = 0; v < 8; ++v) {
            int h = v + 8 * hi;
            float b1v = lds[OFF_B1 + t * 16 + h];
            float w2v = lds[OFF_W2 + t * 16 + h];
            ep += tanhf(acc[v] + b1v) * w2v;
        }
        // combine halves (h=0..7 in lanes 0-15, h=8..9 in lanes 16-31)
        ep = xor16_sum(ep);
        float e = fmaxf(ep + lds[OFF_B2 + t], 0.0f);
        e_rows[rowL * TX + t] = e;   // both halves write identical value
    }

    // ---- Softmax over t and alpha-weighted sum of a (per row) ----
    float* erow = e_rows + rowL * TX;
    float m = -INFINITY;
    for (int t = 0; t < TX; ++t) m = fmaxf(m, erow[t]);
    float sum = 0.0f;
    for (int t = 0; t < TX; ++t) {
        float ex = __expf(erow[t] - m);
        sum += ex;
        erow[t] = ex;                // both halves write identical value
    }
    float inv = 1.0f / sum;

    float4 accO[8];
    #pragma unroll
    for (int j = 0; j < 8; ++j) accO[j] = make_float4(0.f, 0.f, 0.f, 0.f);

    const float* abase = arow + hi * 32;   // lane covers columns [hi*32, hi*32+32)
    for (int t = 0; t < TX; ++t) {
        float alpha = erow[t] * inv;
        const float4* p = (const float4*)(abase + t * DA);
        #pragma unroll
        for (int j = 0; j < 8; ++j) {
            float4 v = p[j];
            accO[j].x += alpha * v.x;
            accO[j].y += alpha * v.y;
            accO[j].z += alpha * v.z;
            accO[j].w += alpha * v.w;
        }
    }

    float4* o = (float4*)(out + (size_t)r * DA + hi * 32);
    #pragma unroll
    for (int j = 0; j < 8; ++j) o[j] = accO[j];
}

extern "C" void kernel_launch(void* const* d_in, const int* in_sizes, int n_in,
                              void* d_out, int out_size, void* d_ws, size_t ws_size,
                              hipStream_t stream) {
    const float* s  = (const float*)d_in[0];
    const float* a  = (const float*)d_in[1];
    const float* W1 = (const float*)d_in[2];
    const float* b1 = (const float*)d_in[3];
    const float* W2 = (const float*)d_in[4];
    const float* b2 = (const float*)d_in[5];
    float* out = (float*)d_out;

    const int nrows  = in_sizes[0] / NS;           // 32768
    const int blocks = nrows / ROWS_PER_BLOCK;     // 256
    const size_t smem = (size_t)LDS_FLOATS * sizeof(float);  // ~273 KB of 320 KB WGP LDS

    attn_wmma_kernel<<<blocks, 256, smem, stream>>>(s, a, W1, b1, W2, b2, out);
}